// NSHELayer_39883066310767
// MI455X (gfx1250) — compile-verified
//
#include <hip/hip_runtime.h>
#include <math.h>

typedef __attribute__((ext_vector_type(2))) float v2f;
typedef __attribute__((ext_vector_type(8))) float v8f;

#define NEG_SLOPE 0.2f

// ---------------------------------------------------------------------------
// Kernel 1: three GEMMs (x@W_r0, x@W_r1, x@loop_weight+bias) via f32 WMMA.
// grid = (N/16 tiles, 3 matrices), block = 256 (8 wave32s).
// Each wave computes a 16x16 tile: rows [blockIdx.x*16, +16), cols [wave*16, +16).
// ---------------------------------------------------------------------------
__global__ __launch_bounds__(256) void gemm3_wmma(
    const float* __restrict__ x, const float* __restrict__ W0,
    const float* __restrict__ W1, const float* __restrict__ Wl,
    const float* __restrict__ bias,
    float* __restrict__ feat0, float* __restrict__ feat1,
    float* __restrict__ out, int N)
{
    __shared__ __align__(16) float xs[16 * 132];   // 16 rows, padded stride 132
    const int tid = threadIdx.x;
    const int rowbase = blockIdx.x * 16;
    const int mat = blockIdx.y;
    const float* __restrict__ W = (mat == 0) ? W0 : ((mat == 1) ? W1 : Wl);
    float* __restrict__ dst = (mat == 0) ? feat0 : ((mat == 1) ? feat1 : out);

    // Stage the 16x128 x tile into LDS (each thread: 8 floats, float4 x2).
    {
        int r = tid >> 4;              // 0..15
        int c = (tid & 15) * 8;        // 0..120
        int grow = rowbase + r;
        float4 v0 = make_float4(0.f, 0.f, 0.f, 0.f), v1 = v0;
        if (grow < N) {
            const float4* p = (const float4*)(x + (size_t)grow * 128 + c);
            v0 = p[0];
            v1 = p[1];
        }
        *(float4*)(xs + r * 132 + c)     = v0;
        *(float4*)(xs + r * 132 + c + 4) = v1;
    }
    __syncthreads();

    const int lane = tid & 31;
    const int wcol = (tid >> 5) * 16;      // wave's column base
    const int col  = wcol + (lane & 15);   // this lane's N column
    const int kh   = (lane >> 4) * 2;      // K sub-offset: 0 (lo half) / 2 (hi half)
    const int m15  = lane & 15;            // A-matrix row for this lane

    v8f acc = {0.f, 0.f, 0.f, 0.f, 0.f, 0.f, 0.f, 0.f};

#pragma unroll
    for (int kk = 0; kk < 32; ++kk) {
        const int k = kk * 4 + kh;
        v2f a;
        a.x = xs[m15 * 132 + k];
        a.y = xs[m15 * 132 + k + 1];
        v2f b;
        b.x = W[(size_t)k * 128 + col];
        b.y = W[(size_t)(k + 1) * 128 + col];
        // 8-arg form: (neg_a, A, neg_b, B, c_mod, C, reuse_a, reuse_b)
        acc = __builtin_amdgcn_wmma_f32_16x16x4_f32(
            false, a, false, b, (short)0, acc, false, false);
    }

    const float bval = (mat == 2) ? bias[col] : 0.f;
#pragma unroll
    for (int v = 0; v < 8; ++v) {
        int row = rowbase + v + ((lane >> 4) << 3);  // D layout: lanes 16-31 -> M+8
        if (row < N) dst[(size_t)row * 128 + col] = acc[v] + bval;
    }
}

// ---------------------------------------------------------------------------
// Kernel 2: per-node attention scalars el[n,h], er[n,h]. One wave per node.
// lane*4 spans the 128 features; head = lane>>3; 8-lane shfl-xor reduction.
// ---------------------------------------------------------------------------
__global__ __launch_bounds__(256) void attn_scalars(
    const float* __restrict__ feat, const float* __restrict__ attn_l,
    const float* __restrict__ attn_r, float* __restrict__ el,
    float* __restrict__ er, int N)
{
    int node = blockIdx.x * (blockDim.x >> 5) + (threadIdx.x >> 5);
    if (node >= N) return;
    int lane = threadIdx.x & 31;
    int h = lane >> 3;
    int c = lane * 4;                                  // == h*32 + (lane&7)*4
    const float4 f = *(const float4*)(feat + (size_t)node * 128 + c);
    const float4 l = *(const float4*)(attn_l + c);
    const float4 r = *(const float4*)(attn_r + c);
    float pl = f.x * l.x + f.y * l.y + f.z * l.z + f.w * l.w;
    float pr = f.x * r.x + f.y * r.y + f.z * r.z + f.w * r.w;
#pragma unroll
    for (int m = 1; m < 8; m <<= 1) {
        pl += __shfl_xor(pl, m, 32);
        pr += __shfl_xor(pr, m, 32);
    }
    if ((lane & 7) == 0) {
        el[node * 4 + h] = pl;
        er[node * 4 + h] = pr;
    }
}

// ---------------------------------------------------------------------------
// Kernel 3: init segment-max (-inf) and segment-sum (0) buffers for both rels.
// ---------------------------------------------------------------------------
__global__ void init_mz(float* m0, float* z0, float* m1, float* z1, int n)
{
    int i = blockIdx.x * blockDim.x + threadIdx.x;
    if (i < n) {
        m0[i] = -INFINITY; z0[i] = 0.f;
        m1[i] = -INFINITY; z1[i] = 0.f;
    }
}

// float atomic-max via signed/unsigned integer atomics (order-preserving).
__device__ __forceinline__ void atomicMaxFloat(float* addr, float val)
{
    if (val >= 0.f)
        atomicMax((int*)addr, __float_as_int(val));
    else
        atomicMin((unsigned int*)addr, __float_as_uint(val));
}

// ---------------------------------------------------------------------------
// Kernel 4: per-(edge,head) leaky-relu score + segment max over dst.
// ---------------------------------------------------------------------------
__global__ void edge_max(const int* __restrict__ src, const int* __restrict__ dst,
                         const float* __restrict__ el, const float* __restrict__ er,
                         float* __restrict__ m, int E)
{
    int t = blockIdx.x * blockDim.x + threadIdx.x;
    if (t >= E * 4) return;
    int e = t >> 2, h = t & 3;
    int s = src[e], d = dst[e];
    float v = el[s * 4 + h] + er[d * 4 + h];
    v = (v > 0.f) ? v : NEG_SLOPE * v;
    atomicMaxFloat(&m[d * 4 + h], v);
}

// ---------------------------------------------------------------------------
// Kernel 5: segment sum of exp(e - m[dst]).
// ---------------------------------------------------------------------------
__global__ void edge_sum(const int* __restrict__ src, const int* __restrict__ dst,
                         const float* __restrict__ el, const float* __restrict__ er,
                         const float* __restrict__ m, float* __restrict__ z, int E)
{
    int t = blockIdx.x * blockDim.x + threadIdx.x;
    if (t >= E * 4) return;
    int e = t >> 2, h = t & 3;
    int s = src[e], d = dst[e];
    float v = el[s * 4 + h] + er[d * 4 + h];
    v = (v > 0.f) ? v : NEG_SLOPE * v;
    atomicAdd(&z[d * 4 + h], expf(v - m[d * 4 + h]));
}

// ---------------------------------------------------------------------------
// Kernel 6: weighted scatter. One wave per edge; lane covers 4 feature cols.
// out[dst] += softmax_coeff(head) * feat[src].
// ---------------------------------------------------------------------------
__global__ __launch_bounds__(256) void edge_scatter(
    const int* __restrict__ src, const int* __restrict__ dst,
    const float* __restrict__ el, const float* __restrict__ er,
    const float* __restrict__ m, const float* __restrict__ z,
    const float* __restrict__ feat, float* __restrict__ out, int E)
{
    int e = blockIdx.x * (blockDim.x >> 5) + (threadIdx.x >> 5);
    if (e >= E) return;
    int lane = threadIdx.x & 31;
    int h = lane >> 3;
    int s = src[e], d = dst[e];
    float v = el[s * 4 + h] + er[d * 4 + h];
    v = (v > 0.f) ? v : NEG_SLOPE * v;
    float a = expf(v - m[d * 4 + h]) / z[d * 4 + h];
    int c = lane * 4;
    const float4 f = *(const float4*)(feat + (size_t)s * 128 + c);
    float* o = out + (size_t)d * 128 + c;
    atomicAdd(o + 0, a * f.x);
    atomicAdd(o + 1, a * f.y);
    atomicAdd(o + 2, a * f.z);
    atomicAdd(o + 3, a * f.w);
}

// ---------------------------------------------------------------------------
// Kernel 7: in-place ReLU on the accumulator (= d_out).
// ---------------------------------------------------------------------------
__global__ void relu_inplace(float* __restrict__ out, int n)
{
    int i = blockIdx.x * blockDim.x + threadIdx.x;
    if (i < n) out[i] = fmaxf(out[i], 0.f);
}

// ---------------------------------------------------------------------------
extern "C" void kernel_launch(void* const* d_in, const int* in_sizes, int n_in,
                              void* d_out, int out_size, void* d_ws, size_t ws_size,
                              hipStream_t stream)
{
    const float* x    = (const float*)d_in[0];
    const int*   src0 = (const int*)d_in[1];
    const int*   dst0 = (const int*)d_in[2];
    const int*   src1 = (const int*)d_in[3];
    const int*   dst1 = (const int*)d_in[4];
    const float* W0   = (const float*)d_in[5];
    const float* al0  = (const float*)d_in[6];
    const float* ar0  = (const float*)d_in[7];
    const float* W1   = (const float*)d_in[8];
    const float* al1  = (const float*)d_in[9];
    const float* ar1  = (const float*)d_in[10];
    const float* Wl   = (const float*)d_in[11];
    const float* bias = (const float*)d_in[12];

    const int N  = in_sizes[0] / 128;
    const int E0 = in_sizes[1];
    const int E1 = in_sizes[3];

    float* ws    = (float*)d_ws;
    float* feat0 = ws;
    float* feat1 = feat0 + (size_t)N * 128;
    float* el0   = feat1 + (size_t)N * 128;
    float* er0   = el0 + (size_t)N * 4;
    float* el1   = er0 + (size_t)N * 4;
    float* er1   = el1 + (size_t)N * 4;
    float* m0    = er1 + (size_t)N * 4;
    float* z0    = m0 + (size_t)N * 4;
    float* m1    = z0 + (size_t)N * 4;
    float* z1    = m1 + (size_t)N * 4;
    float* out   = (float*)d_out;

    // 1) Three WMMA GEMMs (loop GEMM + bias lands directly in d_out).
    gemm3_wmma<<<dim3((N + 15) / 16, 3), 256, 0, stream>>>(
        x, W0, W1, Wl, bias, feat0, feat1, out, N);

    // 2) Per-node attention scalars for both relations.
    attn_scalars<<<(N + 7) / 8, 256, 0, stream>>>(feat0, al0, ar0, el0, er0, N);
    attn_scalars<<<(N + 7) / 8, 256, 0, stream>>>(feat1, al1, ar1, el1, er1, N);

    // 3) Init softmax stats.
    init_mz<<<(N * 4 + 255) / 256, 256, 0, stream>>>(m0, z0, m1, z1, N * 4);

    // 4) Segment max over destinations.
    edge_max<<<(E0 * 4 + 255) / 256, 256, 0, stream>>>(src0, dst0, el0, er0, m0, E0);
    edge_max<<<(E1 * 4 + 255) / 256, 256, 0, stream>>>(src1, dst1, el1, er1, m1, E1);

    // 5) Segment sum of exp(e - m).
    edge_sum<<<(E0 * 4 + 255) / 256, 256, 0, stream>>>(src0, dst0, el0, er0, m0, z0, E0);
    edge_sum<<<(E1 * 4 + 255) / 256, 256, 0, stream>>>(src1, dst1, el1, er1, m1, z1, E1);

    // 6) Weighted scatter into accumulator (d_out already holds x@loop + bias).
    edge_scatter<<<(E0 + 7) / 8, 256, 0, stream>>>(src0, dst0, el0, er0, m0, z0,
                                                   feat0, out, E0);
    edge_scatter<<<(E1 + 7) / 8, 256, 0, stream>>>(src1, dst1, el1, er1, m1, z1,
                                                   feat1, out, E1);

    // 7) Final activation.
    relu_inplace<<<(N * 128 + 255) / 256, 256, 0, stream>>>(out, N * 128);
}